// Topo_22969485099827
// MI455X (gfx1250) — compile-verified
//
#include <hip/hip_runtime.h>
#include <math.h>

typedef __attribute__((ext_vector_type(2))) float v2f;
typedef __attribute__((ext_vector_type(8))) float v8f;

#define B_N  2048
#define T_N  128
#define C_N  3
#define J_N  25
#define E_N  64
#define KPAD 28
#define NPTS 2048

// workspace layout, in float units
#define OFF_MBAR   0        // 2048*25 = 51200
#define OFF_HRAW   51200    // 51200
#define OFF_MINMAX 102400   // 2 (as uint bits)
#define OFF_HPAD   102464   // 2048*28 = 57344
#define OFF_SQ     159808   // 2048
#define OFF_DEATH  161856   // 2047
#define OFF_DMAT   163904   // 2048*2048 = 4194304

// ---- 1) mbar[b,j] = mean over T,C of x[b,t,c,j] ------------------------
__global__ void k_mean(const float* __restrict__ x, float* __restrict__ mbar) {
    int b = blockIdx.x;
    int j = threadIdx.x;
    if (j >= J_N) return;
    const float* xb = x + (size_t)b * (T_N * C_N * J_N);
    float s = 0.f;
    for (int tc = 0; tc < T_N * C_N; ++tc)
        s += xb[tc * J_N + j];                 // 25 consecutive floats per wave step
    mbar[b * J_N + j] = s * (1.0f / (T_N * C_N));
}

// ---- 2) hraw[b,k] = sqrt(sum_j (mbar[b,j]-mbar[b,k])^2) ---------------
__global__ void k_hraw(const float* __restrict__ mbar, float* __restrict__ hraw) {
    __shared__ float m[J_N];
    int b = blockIdx.x, k = threadIdx.x;
    if (k < J_N) m[k] = mbar[b * J_N + k];
    __syncthreads();
    if (k < J_N) {
        float mk = m[k], s = 0.f;
        for (int j = 0; j < J_N; ++j) { float d = m[j] - mk; s += d * d; }
        hraw[b * J_N + k] = sqrtf(s);
    }
}

// ---- 3) global min/max (values are >= 0 so uint bit-order == float order)
__global__ void k_minmax_init(unsigned* mm) { mm[0] = 0x7F800000u; mm[1] = 0u; }

__global__ void k_minmax(const float* __restrict__ h, unsigned* mm, int n) {
    int i = blockIdx.x * blockDim.x + threadIdx.x;
    float lo = (i < n) ? h[i] : INFINITY;
    float hi = (i < n) ? h[i] : 0.0f;
    for (int off = 16; off > 0; off >>= 1) {
        lo = fminf(lo, __shfl_xor(lo, off, 32));
        hi = fmaxf(hi, __shfl_xor(hi, off, 32));
    }
    if ((threadIdx.x & 31) == 0) {
        atomicMin(&mm[0], __float_as_uint(lo));
        atomicMax(&mm[1], __float_as_uint(hi));
    }
}

// ---- 4) normalize, write zero-padded H[2048][28], row sq-norms --------
__global__ void k_normpad(const float* __restrict__ hraw, const unsigned* __restrict__ mm,
                          float* __restrict__ hpad, float* __restrict__ sq) {
    int b = blockIdx.x, j = threadIdx.x;
    float lo  = __uint_as_float(mm[0]);
    float hi  = __uint_as_float(mm[1]);
    float rng = hi - lo;
    bool  safe = rng > 1e-8f;
    float inv  = safe ? 1.0f / rng : 0.0f;
    float v = 0.f;
    if (j < KPAD) {
        if (j < J_N) v = safe ? (hraw[b * J_N + j] - lo) * inv : 0.0f;
        hpad[b * KPAD + j] = v;                // K padding -> exact zeros
    }
    float s = v * v;
    for (int off = 16; off > 0; off >>= 1) s += __shfl_xor(s, off, 32);
    if (j == 0) sq[b] = s;
}

// ---- 5) D = sqrt(max(sq_i + sq_k - 2*H H^T, 1e-12)) via f32 WMMA ------
// One 16x16 tile per wave32; K=28 -> 7 x V_WMMA_F32_16X16X4_F32.
// f32 16x4 A layout: lanes 0-15 hold K=0(vgpr0),1(vgpr1); lanes 16-31 hold K=2,3.
__global__ void __launch_bounds__(128) k_gram(const float* __restrict__ hpad,
                                              const float* __restrict__ sq,
                                              float* __restrict__ dmat) {
    int lane = threadIdx.x & 31;
    int wave = threadIdx.x >> 5;
    int rowBase = (blockIdx.y * 4 + wave) * 16;
    int colBase = blockIdx.x * 16;
    int half = lane >> 4, m = lane & 15;

    const float* arow = hpad + (size_t)(rowBase + m) * KPAD + 2 * half;
    const float* brow = hpad + (size_t)(colBase + m) * KPAD + 2 * half;

    v8f acc = {};
#pragma unroll
    for (int kk = 0; kk < KPAD; kk += 4) {
        v2f a, b;
        a.x = arow[kk]; a.y = arow[kk + 1];
        b.x = brow[kk]; b.y = brow[kk + 1];
        acc = __builtin_amdgcn_wmma_f32_16x16x4_f32(
            /*neg_a=*/false, a, /*neg_b=*/false, b,
            /*c_mod=*/(short)0, acc, /*reuse_a=*/false, /*reuse_b=*/false);
    }
#pragma unroll
    for (int r = 0; r < 8; ++r) {            // C/D: vgpr r -> M = r (+8 for hi half)
        int M = rowBase + r + 8 * half;
        int N = colBase + m;
        float d2 = sq[M] + sq[N] - 2.0f * acc[r];
        dmat[(size_t)M * NPTS + N] = sqrtf(fmaxf(d2, 1e-12f));
    }
}

// ---- 6) Prim's MST: single workgroup, 2047 sequential steps -----------
__global__ void __launch_bounds__(1024) k_mst(const float* __restrict__ dmat,
                                              float* __restrict__ deaths) {
    __shared__ float sv[32];
    __shared__ int   si[32];
    __shared__ int   sj;
    int t = threadIdx.x;
    int lane = t & 31, wave = t >> 5;
    int i1 = t, i2 = t + 1024;
    float m1 = dmat[i1];                     // Dm[0][i]
    float m2 = dmat[i2];
    bool in1 = (i1 == 0), in2 = false;

    for (int step = 0; step < NPTS - 1; ++step) {
        float v1 = in1 ? INFINITY : m1;
        float v2 = in2 ? INFINITY : m2;
        float v;  int idx;
        if (v2 < v1) { v = v2; idx = i2; } else { v = v1; idx = i1; }
        for (int off = 16; off > 0; off >>= 1) {       // in-wave (val,idx) argmin
            float ov = __shfl_xor(v,   off, 32);
            int   oi = __shfl_xor(idx, off, 32);
            if (ov < v || (ov == v && oi < idx)) { v = ov; idx = oi; }
        }
        if (lane == 0) { sv[wave] = v; si[wave] = idx; }
        __syncthreads();
        if (wave == 0) {                               // reduce 32 wave winners
            float wv = sv[lane]; int wi = si[lane];
            for (int off = 16; off > 0; off >>= 1) {
                float ov = __shfl_xor(wv, off, 32);
                int   oi = __shfl_xor(wi, off, 32);
                if (ov < wv || (ov == wv && oi < wi)) { wv = ov; wi = oi; }
            }
            if (lane == 0) { sj = wi; deaths[step] = wv; }
        }
        __syncthreads();
        int j = sj;
        if (i1 == j) in1 = true;
        if (i2 == j) in2 = true;
        const float* row = dmat + (size_t)j * NPTS;    // 8KB coalesced row from L2
        m1 = fminf(m1, row[i1]);
        m2 = fminf(m2, row[i2]);
    }
}

// ---- 7) structure elements: out[e] = sum_p exp(-(s0^2 c0^2 + s1^2 (d_p-c1)^2))
__global__ void k_sel(const float* __restrict__ deaths,
                      const float* __restrict__ centres,
                      const float* __restrict__ sharp,
                      float* __restrict__ out) {
    int e = threadIdx.x;
    if (e >= E_N) return;
    float c0 = centres[e * 2 + 0], c1 = centres[e * 2 + 1];
    float s0 = sharp[e * 2 + 0],   s1 = sharp[e * 2 + 1];
    float t0 = s0 * s0 * c0 * c0;            // birth coord is always 0
    float s1sq = s1 * s1;
    float acc = 0.f;
    for (int p = 0; p < NPTS - 1; ++p) {
        float d = deaths[p] - c1;            // broadcast read across all lanes
        acc += expf(-(t0 + s1sq * d * d));
    }
    out[e] = acc;
}

extern "C" void kernel_launch(void* const* d_in, const int* in_sizes, int n_in,
                              void* d_out, int out_size, void* d_ws, size_t ws_size,
                              hipStream_t stream) {
    const float* x       = (const float*)d_in[0];   // [2048,128,3,25]
    const float* centres = (const float*)d_in[1];   // [64,2]
    const float* sharp   = (const float*)d_in[2];   // [64,2]
    float* ws  = (float*)d_ws;
    float* out = (float*)d_out;                     // [1,64] f32

    float*    mbar  = ws + OFF_MBAR;
    float*    hraw  = ws + OFF_HRAW;
    unsigned* mm    = (unsigned*)(ws + OFF_MINMAX);
    float*    hpad  = ws + OFF_HPAD;
    float*    sqn   = ws + OFF_SQ;
    float*    death = ws + OFF_DEATH;
    float*    dmat  = ws + OFF_DMAT;

    k_mean       <<<B_N, 32, 0, stream>>>(x, mbar);
    k_hraw       <<<B_N, 32, 0, stream>>>(mbar, hraw);
    k_minmax_init<<<1, 1, 0, stream>>>(mm);
    k_minmax     <<<(B_N * J_N + 255) / 256, 256, 0, stream>>>(hraw, mm, B_N * J_N);
    k_normpad    <<<B_N, 32, 0, stream>>>(hraw, mm, hpad, sqn);
    dim3 ggrid(NPTS / 16, NPTS / 16 / 4);           // 128 x 32, 4 waves/block
    k_gram       <<<ggrid, 128, 0, stream>>>(hpad, sqn, dmat);
    k_mst        <<<1, 1024, 0, stream>>>(dmat, death);
    k_sel        <<<1, E_N, 0, stream>>>(death, centres, sharp, out);
}